// PCGraphConv_83537113907851
// MI455X (gfx1250) — compile-verified
//
#include <hip/hip_runtime.h>

#define B      8
#define CHUNK  1024
#define TPB    256
#define EDGE_WGS 512

typedef __attribute__((ext_vector_type(4))) unsigned int v4u;
typedef __attribute__((ext_vector_type(8))) int          v8i;
typedef __attribute__((ext_vector_type(4))) int          v4i;

// LDS byte offset of a __shared__ object: low 32 bits of the generic address.
__device__ __forceinline__ unsigned lds_off(const void* p) {
  return (unsigned)(unsigned long long)p;
}

// 1-D TDM copy: nelem elements of (1<<dsz_log2) bytes, global -> LDS.
// D# packed per CDNA5 ISA 8.3/8.4 (group0: count=1,lds_addr,global_addr,type=2;
// group1: data_size, tensor_dim0=tile_dim0=nelem, tile_dim1=0 => 1-D tile).
__device__ __forceinline__ void tdm_load_1d(unsigned lds_addr, const void* gptr,
                                            unsigned nelem, unsigned dsz_log2) {
  unsigned long long ga = (unsigned long long)gptr;
  v4u g0;
  g0.x = 1u;                                               // count=1, user mode
  g0.y = lds_addr;                                         // lds_addr[31:0]
  g0.z = (unsigned)(ga & 0xFFFFFFFFull);                   // global_addr[31:0]
  g0.w = (unsigned)((ga >> 32) & 0x1FFFFFFull) | (2u << 30); // addr[56:32] | type=2
  v8i g1;
  g1[0] = (int)(dsz_log2 << 16);                           // wg_mask=0, data_size
  g1[1] = (int)((nelem & 0xFFFFu) << 16);                  // tensor_dim0[15:0]
  g1[2] = (int)(((nelem >> 16) & 0xFFFFu) | (1u << 16));   // tdim0[31:16], tensor_dim1=1
  g1[3] = (int)((nelem & 0xFFFFu) << 16);                  // tile_dim0 = nelem
  g1[4] = 0;                                               // tile_dim1=0, tile_dim2=0
  g1[5] = (int)nelem;                                      // tensor_dim0_stride[31:0]
  g1[6] = 0;
  g1[7] = 0;
  v4i z4 = {0, 0, 0, 0};
  v8i z8 = {0, 0, 0, 0, 0, 0, 0, 0};
  __builtin_amdgcn_tensor_load_to_lds(g0, g1, z4, z4, z8, 0);
}

__global__ void node_init(const float* __restrict__ vals, float* __restrict__ fx,
                          float* __restrict__ aggr, float* __restrict__ pred, int BN) {
  int i = blockIdx.x * blockDim.x + threadIdx.x;
  if (i < BN) {
    fx[i]   = tanhf(vals[i]);
    pred[i] = 0.0f;
    aggr[i] = 0.0f;
  }
}

__global__ void node_err(const float* __restrict__ vals, const float* __restrict__ pred,
                         float* __restrict__ err, int BN) {
  int i = blockIdx.x * blockDim.x + threadIdx.x;
  if (i < BN) err[i] = vals[i] - pred[i];
}

__global__ void node_dx(const float* __restrict__ fx, const float* __restrict__ err,
                        const float* __restrict__ aggr, float* __restrict__ dx, int BN) {
  int i = blockIdx.x * blockDim.x + threadIdx.x;
  if (i < BN) {
    float f = fx[i];
    dx[i] = err[i] - (1.0f - f * f) * aggr[i];
  }
}

// One kernel for both edge passes.
// swap_roles==0: scatter[b*N+tgt] += w * gather[b*N+src]   (prediction pass)
// swap_roles==1: scatter[b*N+src] += w * gather[b*N+tgt]   (value pass)
__global__ __launch_bounds__(TPB) void edge_scatter(
    const float* __restrict__ w, const long long* __restrict__ ei,
    const float* __restrict__ gather, float* __restrict__ scatter,
    int E, int N, int edges_per_wg, int swap_roles) {
  __shared__ float    lw[2][CHUNK];        //  8 KB
  __shared__ unsigned ls[2][2 * CHUNK];    // 16 KB (raw int64 dwords)
  __shared__ unsigned lt[2][2 * CHUNK];    // 16 KB

  int base = blockIdx.x * edges_per_wg;
  if (base >= E) return;                    // uniform per block
  int span = min(edges_per_wg, E - base);
  int nch  = (span + CHUNK - 1) / CHUNK;

  const bool issuer = (threadIdx.x < 32);   // wave 0 only (wave-uniform)

  auto issue = [&](int c, int buf) {
    int e0 = base + c * CHUNK;
    unsigned len = (unsigned)min(CHUNK, span - c * CHUNK);
    tdm_load_1d(lds_off(&lw[buf][0]), w + e0, len, 2);                    // f32
    tdm_load_1d(lds_off(&ls[buf][0]), ei + e0, len, 3);                   // i64 src
    tdm_load_1d(lds_off(&lt[buf][0]), ei + (size_t)E + e0, len, 3);       // i64 tgt
  };

  if (issuer) issue(0, 0);

  int buf = 0;
  for (int c = 0; c < nch; ++c) {
    bool more = (c + 1 < nch);
    if (issuer) {
      if (more) {
        issue(c + 1, buf ^ 1);                      // prefetch next chunk
        __builtin_amdgcn_s_wait_tensorcnt(3);       // current chunk's 3 loads done
      } else {
        __builtin_amdgcn_s_wait_tensorcnt(0);
      }
    }
    __syncthreads();                                // publish LDS tile to all waves

    // Explicit DS-pipe reads of the TDM-written tile (compiler cannot see the
    // TDM writes, so keep these as pinned ds_load_b32 + one dscnt wait).
    unsigned base_w = lds_off(&lw[buf][0]);
    unsigned base_s = lds_off(&ls[buf][0]);
    unsigned base_t = lds_off(&lt[buf][0]);

    int len = min(CHUNK, span - c * CHUNK);
    for (int i = threadIdx.x; i < len; i += TPB) {
      unsigned wt_bits, s_u, t_u;
      asm volatile(
          "ds_load_b32 %0, %3\n\t"
          "ds_load_b32 %1, %4\n\t"
          "ds_load_b32 %2, %5\n\t"
          "s_wait_dscnt 0x0"
          : "=&v"(wt_bits), "=&v"(s_u), "=&v"(t_u)
          : "v"(base_w + 4u * (unsigned)i),
            "v"(base_s + 8u * (unsigned)i),
            "v"(base_t + 8u * (unsigned)i)
          : "memory");
      float wt = __uint_as_float(wt_bits);
      int s = (int)s_u;                             // low dword of int64 (< N)
      int t = (int)t_u;
      int gi = swap_roles ? t : s;
      int si = swap_roles ? s : t;
      const float* gp = gather + gi;
      float*       sp = scatter + si;
#pragma unroll
      for (int b = 0; b < B; ++b)
        atomicAdd(sp + b * N, wt * gp[b * N]);
    }
    __syncthreads();                                // buffer reusable next iter
    buf ^= 1;
  }
}

extern "C" void kernel_launch(void* const* d_in, const int* in_sizes, int n_in,
                              void* d_out, int out_size, void* d_ws, size_t ws_size,
                              hipStream_t stream) {
  const float*     vals = (const float*)d_in[0];      // [B*N] f32
  const float*     wts  = (const float*)d_in[1];      // [E]   f32
  const long long* ei   = (const long long*)d_in[2];  // [2,E] int64

  int BN = in_sizes[0];
  int E  = in_sizes[1];
  int N  = BN / B;

  float* out  = (float*)d_out;
  float* pred = out;            // [BN]
  float* err  = out + BN;       // [BN]
  float* dx   = out + 2 * BN;   // [BN]
  float* fx   = (float*)d_ws;   // [BN]
  float* aggr = fx + BN;        // [BN]

  int nb = (BN + TPB - 1) / TPB;
  node_init<<<nb, TPB, 0, stream>>>(vals, fx, aggr, pred, BN);

  int epw = (E + EDGE_WGS - 1) / EDGE_WGS;
  edge_scatter<<<EDGE_WGS, TPB, 0, stream>>>(wts, ei, fx, pred, E, N, epw, 0);
  node_err<<<nb, TPB, 0, stream>>>(vals, pred, err, BN);
  edge_scatter<<<EDGE_WGS, TPB, 0, stream>>>(wts, ei, err, aggr, E, N, epw, 1);
  node_dx<<<nb, TPB, 0, stream>>>(fx, err, aggr, dx, BN);
}